// segment3_77610059039206
// MI455X (gfx1250) — compile-verified
//
#include <hip/hip_runtime.h>
#include <hip/hip_bf16.h>
#include <math.h>

typedef __attribute__((ext_vector_type(2))) float v2f;
typedef __attribute__((ext_vector_type(8))) float v8f;

#if defined(__has_builtin)
#if __has_builtin(__builtin_amdgcn_wmma_f32_16x16x4_f32)
#define HAVE_WMMA_F32X4 1
#endif
#if __has_builtin(__builtin_amdgcn_update_dpp)
#define HAVE_DPP 1
#endif
#endif

#define D 64
#define TILE 16
#define EMB_STRIDE 68   // padded LDS row stride (dwords) for the A tile

#if HAVE_DPP
// butterfly add step across lanes within a 16-lane DPP row (pure VALU, no LDS)
template <int CTRL>
__device__ __forceinline__ float dpp_add(float x) {
    int y = __builtin_amdgcn_update_dpp(0, __builtin_bit_cast(int, x),
                                        CTRL, 0xF, 0xF, true);
    return x + __builtin_bit_cast(float, y);
}
#endif

// ---------------- bucketing kernels ----------------

__global__ void init_zero_kernel(int* p, int n) {
    int i = blockIdx.x * blockDim.x + threadIdx.x;
    if (i < n) p[i] = 0;
}

__global__ void hist_kernel(const int* __restrict__ exp_ind, int* __restrict__ counts, int n) {
    int i = blockIdx.x * blockDim.x + threadIdx.x;
    if (i < n) atomicAdd(&counts[exp_ind[i]], 1);
}

// single-block exclusive scan (Hillis-Steele over 1024-chunks with carry)
__global__ __launch_bounds__(1024)
void scan_kernel(const int* __restrict__ counts, int* __restrict__ offsets, int n) {
    __shared__ int tmp[1024];
    __shared__ int carry;
    int tid = threadIdx.x;
    if (tid == 0) carry = 0;
    __syncthreads();
    for (int base = 0; base < n; base += 1024) {
        int idx = base + tid;
        int v = (idx < n) ? counts[idx] : 0;
        tmp[tid] = v;
        __syncthreads();
        #pragma unroll
        for (int off = 1; off < 1024; off <<= 1) {
            int t = (tid >= off) ? tmp[tid - off] : 0;
            __syncthreads();
            tmp[tid] += t;
            __syncthreads();
        }
        int incl = tmp[tid];
        if (idx < n) offsets[idx] = carry + incl - v;   // exclusive
        __syncthreads();
        if (tid == 0) carry += tmp[1023];
        __syncthreads();
    }
}

__global__ void scatter_kernel(const int* __restrict__ exp_ind,
                               const int* __restrict__ offsets,
                               int* __restrict__ cursors,
                               int* __restrict__ bucket, int n) {
    int i = blockIdx.x * blockDim.x + threadIdx.x;
    if (i < n) {
        int u = exp_ind[i];
        int pos = atomicAdd(&cursors[u], 1);
        bucket[offsets[u] + pos] = i;
    }
}

// ---------------- per-user WMMA GEMM ----------------
// one workgroup (128 threads = 4 waves) per user.
// (I + morph) staged in LDS once; B fragments hoisted into registers once;
// the item-tile loop does only A-fragment LDS loads + 16 WMMAs; row norms
// via a pure-VALU DPP butterfly; normalized rows stored from registers.

__global__ __launch_bounds__(128)
void user_gemm_kernel(const int* __restrict__ items,
                      const float* __restrict__ item_emb,
                      const float* __restrict__ seg2,
                      const int* __restrict__ offsets,
                      const int* __restrict__ counts,
                      const int* __restrict__ bucket,
                      float* __restrict__ out) {
    const int user = blockIdx.x;
    const int cnt  = counts[user];
    if (cnt == 0) return;                 // uniform per block
    const int base = offsets[user];

    __shared__ float sM[D * D];             // (I + morph) [d][k], 16 KB
    __shared__ float sA[TILE * EMB_STRIDE]; // emb tile, padded rows
    __shared__ float sP[4 * TILE];          // [wave][row] partial sum-of-squares
    __shared__ float sInv[TILE];
    __shared__ int   sRow[TILE];            // original item index per row (-1 pad)

    const int tid  = threadIdx.x;         // 0..127
    const int lane = tid & 31;
    const int wave = tid >> 5;            // 0..3
    const int n0   = wave * 16;           // this wave's N-tile
    const int hi   = lane >> 4;           // 0/1 half-wave
    const int l15  = lane & 15;

    // stage morph: 4096 floats = 1024 float4, 8 per thread (coalesced, L2-friendly)
    {
        const float4* gm  = (const float4*)(seg2 + (size_t)user * (D * D));
        float4*       sm4 = (float4*)sM;
        #pragma unroll
        for (int i = 0; i < 8; ++i) sm4[tid + i * 128] = gm[tid + i * 128];
    }
    __syncthreads();
    // fold "+emb" bias into the GEMM: emb + emb@M == emb @ (I + M)
    if (tid < D) sM[tid * D + tid] += 1.0f;
    __syncthreads();

    // hoist B fragments for this wave's N-tile out of the item loop:
    // b[kk/4] = (B[ka][n], B[ka+1][n]) with ka = kk + 2*hi, n = n0 + l15
    v2f bfrag[16];
    #pragma unroll
    for (int kk = 0; kk < D; kk += 4) {
        int ka = kk + 2 * hi;
        bfrag[kk >> 2].x = sM[ka * D + n0 + l15];
        bfrag[kk >> 2].y = sM[(ka + 1) * D + n0 + l15];
    }

    for (int t0 = 0; t0 < cnt; t0 += TILE) {
        // ---- stage 16 gathered embedding rows into sA (zero-pad tail) ----
        {
            int r = tid >> 3, seg = tid & 7;      // 8 threads per row, 8 floats each
            int idx = t0 + r;
            int orig = (idx < cnt) ? bucket[base + idx] : -1;
            if (seg == 0) sRow[r] = orig;
            float4 a0 = make_float4(0.f, 0.f, 0.f, 0.f), a1 = a0;
            if (orig >= 0) {
                const float4* ge = (const float4*)(item_emb + (size_t)items[orig] * D);
                a0 = ge[seg * 2 + 0];
                a1 = ge[seg * 2 + 1];
            }
            float* dst = sA + r * EMB_STRIDE + seg * 8;
            *(float4*)(dst + 0) = a0;
            *(float4*)(dst + 4) = a1;
        }
        __syncthreads();

        // ---- C(16x16) = A(16x64) * (I+M)(64x64)[:, n0:n0+16], 16 WMMA steps ----
        v8f c = {};
        #pragma unroll
        for (int kk = 0; kk < D; kk += 4) {
            int ka = kk + 2 * hi;
            v2f a;
            a.x = sA[l15 * EMB_STRIDE + ka];          // A[m=l15][ka]
            a.y = sA[l15 * EMB_STRIDE + ka + 1];
#if HAVE_WMMA_F32X4
            c = __builtin_amdgcn_wmma_f32_16x16x4_f32(
                    false, a, false, bfrag[kk >> 2], (short)0, c, false, false);
#else
            // scalar fallback (same fragment layout)
            #pragma unroll
            for (int r = 0; r < 8; ++r) {
                int m = r + 8 * hi;
                float acc = c[r];
                #pragma unroll
                for (int k = 0; k < 4; ++k)
                    acc += sA[m * EMB_STRIDE + kk + k] * sM[(kk + k) * D + n0 + l15];
                c[r] = acc;
            }
#endif
        }

        // ---- row sum-of-squares across this wave's 16 columns.
        //      C/D layout: VGPR r holds (M = r + 8*hi, N = n0 + l15); reduce
        //      over l15 within each 16-lane DPP row: xor1,xor2 quad_perms,
        //      then half_mirror (quad^1) and mirror (quad^3). Pure VALU. ----
        float srow[8];
        #pragma unroll
        for (int r = 0; r < 8; ++r) {
            float s = c[r] * c[r];
#if HAVE_DPP
            s = dpp_add<0x0B1>(s);   // quad_perm(1,0,3,2): + lane^1
            s = dpp_add<0x04E>(s);   // quad_perm(2,3,0,1): + lane^2
            s = dpp_add<0x141>(s);   // row_half_mirror:    + quad^1
            s = dpp_add<0x140>(s);   // row_mirror:         + quad^3
#else
            s += __shfl_xor(s, 1);
            s += __shfl_xor(s, 2);
            s += __shfl_xor(s, 4);
            s += __shfl_xor(s, 8);
#endif
            srow[r] = s;             // every lane now holds its row's partial
        }
        if (l15 == 0) {              // one writer per half-wave; rows contiguous
            float4* dst = (float4*)&sP[wave * TILE + 8 * hi];
            dst[0] = make_float4(srow[0], srow[1], srow[2], srow[3]);
            dst[1] = make_float4(srow[4], srow[5], srow[6], srow[7]);
        }
        __syncthreads();

        if (tid < TILE) {
            float s = sP[0 * TILE + tid] + sP[1 * TILE + tid]
                    + sP[2 * TILE + tid] + sP[3 * TILE + tid];
            sInv[tid] = 1.0f / fmaxf(sqrtf(s), 1e-12f);
        }
        __syncthreads();

        // ---- normalized rows straight from registers to out[orig_item] ----
        {
            const int4*   rp = (const int4*)&sRow[8 * hi];
            const float4* ip = (const float4*)&sInv[8 * hi];
            int4   r0 = rp[0], r1 = rp[1];
            float4 i0 = ip[0], i1 = ip[1];
            int   orig_m[8] = { r0.x, r0.y, r0.z, r0.w, r1.x, r1.y, r1.z, r1.w };
            float inv_m[8]  = { i0.x, i0.y, i0.z, i0.w, i1.x, i1.y, i1.z, i1.w };
            #pragma unroll
            for (int r = 0; r < 8; ++r) {
                if (orig_m[r] >= 0)
                    out[(size_t)orig_m[r] * D + n0 + l15] = c[r] * inv_m[r];
            }
        }
        __syncthreads();                              // before next tile reuses sA
    }
}

// ---------------- naive fallback (workspace too small) ----------------

__global__ __launch_bounds__(64)
void naive_kernel(const int* __restrict__ items, const int* __restrict__ exp_ind,
                  const float* __restrict__ item_emb, const float* __restrict__ seg2,
                  float* __restrict__ out, int n) {
    int i = blockIdx.x;
    if (i >= n) return;
    int k = threadIdx.x;
    __shared__ float se[D];
    __shared__ float red[D];
    se[k] = item_emb[(size_t)items[i] * D + k];
    __syncthreads();
    const float* M = seg2 + (size_t)exp_ind[i] * (D * D);
    float acc = se[k];
    #pragma unroll 8
    for (int d = 0; d < D; ++d) acc += M[d * D + k] * se[d];
    red[k] = acc * acc;
    __syncthreads();
    #pragma unroll
    for (int s = 32; s > 0; s >>= 1) {
        if (k < s) red[k] += red[k + s];
        __syncthreads();
    }
    out[(size_t)i * D + k] = acc / fmaxf(sqrtf(red[0]), 1e-12f);
}

// ---------------- launch ----------------

extern "C" void kernel_launch(void* const* d_in, const int* in_sizes, int n_in,
                              void* d_out, int out_size, void* d_ws, size_t ws_size,
                              hipStream_t stream) {
    const int*   items    = (const int*)d_in[0];
    const int*   exp_ind  = (const int*)d_in[1];
    const float* item_emb = (const float*)d_in[2];
    const float* seg2     = (const float*)d_in[3];
    float*       out      = (float*)d_out;

    const int N       = in_sizes[0];
    const int n_users = in_sizes[3] / (D * D);

    size_t need = (size_t)(3 * n_users + N) * sizeof(int);
    if (n_users <= 0 || ws_size < need) {
        naive_kernel<<<N, 64, 0, stream>>>(items, exp_ind, item_emb, seg2, out, N);
        return;
    }

    int* counts  = (int*)d_ws;
    int* cursors = counts + n_users;     // contiguous with counts for joint zero-init
    int* offsets = cursors + n_users;
    int* bucket  = offsets + n_users;

    init_zero_kernel<<<(2 * n_users + 255) / 256, 256, 0, stream>>>(counts, 2 * n_users);
    hist_kernel<<<(N + 255) / 256, 256, 0, stream>>>(exp_ind, counts, N);
    scan_kernel<<<1, 1024, 0, stream>>>(counts, offsets, n_users);
    scatter_kernel<<<(N + 255) / 256, 256, 0, stream>>>(exp_ind, offsets, cursors, bucket, N);
    user_gemm_kernel<<<n_users, 128, 0, stream>>>(items, item_emb, seg2,
                                                  offsets, counts, bucket, out);
}